// TwoLayerRGCN_22514218566018
// MI455X (gfx1250) — compile-verified
//
#include <hip/hip_runtime.h>
#include <hip/hip_bf16.h>

typedef __attribute__((ext_vector_type(2))) float v2f;
typedef __attribute__((ext_vector_type(8))) float v8f;

#define DFEAT 128
#define NREL  8

// ---------------------------------------------------------------------------
// Edge-count kernel: cnt[r*N + dst] += 1  (float so we can reuse fp atomics)
// ---------------------------------------------------------------------------
__global__ void rgcn_count(const int* __restrict__ dst,
                           const int* __restrict__ et,
                           float* __restrict__ cnt, int E, int N) {
    int e = blockIdx.x * blockDim.x + threadIdx.x;
    if (e < E) {
        int seg = et[e] * N + dst[e];
        unsafeAtomicAdd(&cnt[seg], 1.0f);
    }
}

// cnt -> 1 / max(cnt, 1)
__global__ void rgcn_recip(float* __restrict__ cnt, int n) {
    int i = blockIdx.x * blockDim.x + threadIdx.x;
    if (i < n) cnt[i] = 1.0f / fmaxf(cnt[i], 1.0f);
}

// ---------------------------------------------------------------------------
// Scatter: one wave32 per edge; lane l moves feat[src, 4l..4l+3] into
// agg[(r*N + dst)*128 + 4l ...] with hardware f32 atomics.
// ---------------------------------------------------------------------------
__global__ void rgcn_scatter(const float* __restrict__ feat,
                             const int* __restrict__ src,
                             const int* __restrict__ dst,
                             const int* __restrict__ et,
                             float* __restrict__ agg, int E, int N) {
    int gw   = (blockIdx.x * blockDim.x + threadIdx.x) >> 5;
    int lane = threadIdx.x & 31;
    int nw   = (gridDim.x * blockDim.x) >> 5;
    for (int e = gw; e < E; e += nw) {
        int s = src[e];
        int d = dst[e];
        int r = et[e];
        const float4 v = *(const float4*)(feat + (size_t)s * DFEAT + lane * 4);
        float* ap = agg + ((size_t)r * N + d) * DFEAT + lane * 4;
        unsafeAtomicAdd(ap + 0, v.x);
        unsafeAtomicAdd(ap + 1, v.y);
        unsafeAtomicAdd(ap + 2, v.z);
        unsafeAtomicAdd(ap + 3, v.w);
    }
}

// ---------------------------------------------------------------------------
// Fused RGCN GEMM (fp32 WMMA):
//   out[n, :] = act( sum_r (agg[r,n,:] * rcnt[r,n]) @ W[r] + feat[n,:] @ root + bias )
// One wave computes a 16x16 output tile with V_WMMA_F32_16X16X4_F32,
// looping K over 9 chunks of 128 (8 relations + root).
// Grid: N/16 blocks; block = (HOUT/16) waves. EXEC stays all-ones.
// ---------------------------------------------------------------------------
template <int HOUT, bool RELU>
__global__ __launch_bounds__(32 * (HOUT / 16))
void rgcn_gemm(const float* __restrict__ agg,   // [R, N, 128]
               const float* __restrict__ rcnt,  // [R, N]  (reciprocal counts)
               const float* __restrict__ feat,  // [N, 128]
               const float* __restrict__ W,     // [R, 128, HOUT]
               const float* __restrict__ root,  // [128, HOUT]
               const float* __restrict__ bias,  // [HOUT]
               float* __restrict__ out,         // [N, HOUT]
               int N) {
    const int wave = threadIdx.x >> 5;
    const int lane = threadIdx.x & 31;
    const int row0 = blockIdx.x * 16;
    const int col0 = wave * 16;
    const int half = lane >> 4;   // 0: K pair {0,1}, 1: K pair {2,3}
    const int l15  = lane & 15;
    const int koff = half * 2;

    v8f acc = {};

    for (int r = 0; r < NREL + 1; ++r) {
        const float* Achunk;
        const float* Bchunk;
        float scale;
        if (r < NREL) {
            Achunk = agg + ((size_t)r * N + row0) * DFEAT;
            Bchunk = W + (size_t)r * DFEAT * HOUT;
            scale  = rcnt[(size_t)r * N + row0 + l15];   // per-row mean scale
        } else {
            Achunk = feat + (size_t)row0 * DFEAT;
            Bchunk = root;
            scale  = 1.0f;
        }
        // A fragment source: row (l15) of the 16-row tile, K offset koff
        const float* arow = Achunk + (size_t)l15 * DFEAT + koff;
        // B fragment source: column (col0+l15), K offset koff
        const float* bcol = Bchunk + (size_t)koff * HOUT + col0 + l15;

#pragma unroll 8
        for (int kk = 0; kk < DFEAT; kk += 4) {
            v2f a, b;
            a.x = arow[kk + 0] * scale;
            a.y = arow[kk + 1] * scale;
            b.x = bcol[(size_t)kk * HOUT];
            b.y = bcol[(size_t)kk * HOUT + HOUT];
            acc = __builtin_amdgcn_wmma_f32_16x16x4_f32(
                /*neg_a=*/false, a, /*neg_b=*/false, b,
                /*c_mod=*/(short)0, acc, /*reuse_a=*/false, /*reuse_b=*/false);
        }
    }

    // Epilogue: C/D layout -> lane l15 = column, VGPR v = row (v + 8*half)
    const int col = col0 + l15;
    const float bv = bias[col];
#pragma unroll
    for (int v = 0; v < 8; ++v) {
        int row = row0 + v + half * 8;
        float val = acc[v] + bv;
        if (RELU) val = fmaxf(val, 0.0f);
        out[(size_t)row * HOUT + col] = val;
    }
}

// ---------------------------------------------------------------------------
// Launch
// ---------------------------------------------------------------------------
extern "C" void kernel_launch(void* const* d_in, const int* in_sizes, int n_in,
                              void* d_out, int out_size, void* d_ws, size_t ws_size,
                              hipStream_t stream) {
    const float* x     = (const float*)d_in[0];
    const int*   ei    = (const int*)d_in[1];   // [2, E]
    const int*   et    = (const int*)d_in[2];   // [E]
    const float* W1    = (const float*)d_in[3];
    const float* root1 = (const float*)d_in[4];
    const float* bias1 = (const float*)d_in[5];
    const float* W2    = (const float*)d_in[6];
    const float* root2 = (const float*)d_in[7];
    const float* bias2 = (const float*)d_in[8];

    const int N = in_sizes[0] / DFEAT;   // 50000
    const int E = in_sizes[1] / 2;       // 800000
    const int* src = ei;
    const int* dst = ei + E;

    char* ws = (char*)d_ws;
    const size_t aggBytes = (size_t)NREL * N * DFEAT * sizeof(float); // 204.8 MB
    const size_t cntBytes = (size_t)NREL * N * sizeof(float);         //   1.6 MB
    float* agg = (float*)ws;
    float* cnt = (float*)(ws + aggBytes);
    float* h   = (float*)(ws + aggBytes + cntBytes);                  // [N, 128]

    // ---- counts (shared by both layers) + zero agg ----
    hipMemsetAsync(agg, 0, aggBytes + cntBytes, stream);
    rgcn_count<<<(E + 255) / 256, 256, 0, stream>>>(dst, et, cnt, E, N);
    rgcn_recip<<<(NREL * N + 255) / 256, 256, 0, stream>>>(cnt, NREL * N);

    // ---- layer 1 ----
    rgcn_scatter<<<12500, 256, 0, stream>>>(x, src, dst, et, agg, E, N);
    rgcn_gemm<128, true><<<N / 16, 256, 0, stream>>>(agg, cnt, x, W1, root1, bias1, h, N);

    // ---- layer 2 ----
    hipMemsetAsync(agg, 0, aggBytes, stream);
    rgcn_scatter<<<12500, 256, 0, stream>>>(h, src, dst, et, agg, E, N);
    rgcn_gemm<64, false><<<N / 16, 128, 0, stream>>>(agg, cnt, h, W2, root2, bias2,
                                                     (float*)d_out, N);
}